// FPNSemanticHeadDPCR_43035572306077
// MI455X (gfx1250) — compile-verified
//
#include <hip/hip_runtime.h>

#define H   64
#define W   192
#define HP  66
#define WP  194
#define PP  (HP*WP)     // 12804 padded spatial size
#define HW  (H*W)       // 12288
#define CIN 256
#define COUT 128

typedef __attribute__((ext_vector_type(2))) float v2f;
typedef __attribute__((ext_vector_type(8))) float v8f;

// ---------------------------------------------------------------------------
// Kernel 1: deformable 3x3 gather + range_out contraction -> y0 (256 x 12288)
// ---------------------------------------------------------------------------
__device__ __forceinline__ int remap_pad(float qf, float& wt) {
    // qf is an integer-valued padded index in [0, PP-1]; map to unpadded x
    // index, or zero the weight if it lands in the zero-pad border.
    int q  = (int)qf;
    int hp = q / WP;
    int wp = q - hp * WP;
    if (hp < 1 || hp > H || wp < 1 || wp > W) { wt = 0.0f; return 0; }
    return (hp - 1) * W + (wp - 1);
}

__global__ void deform_gather_kernel(const float* __restrict__ x,     // (256,64,192)
                                     const float* __restrict__ xr,    // (64,192)
                                     const float* __restrict__ rout,  // (256,9)
                                     float* __restrict__ y0)          // (256,12288)
{
    int p = blockIdx.x * blockDim.x + threadIdx.x;   // pixel
    if (p >= HW) return;
    int c0 = blockIdx.y * 32;                        // channel slice
    int h = p / W, w = p - h * W;

    float o  = 3.0f / (1.0f + __expf(-xr[p]));       // 3*sigmoid
    float v0 = (float)((h + 1) * WP + (w + 1));

    const float xoff[9] = {-1,0,1,-1,0,1,-1,0,1};
    const float yoff[9] = {-1,-1,-1,0,0,0,1,1,1};

    int   iF[9], iF1[9], iC[9], iC1[9];
    float wF[9], wF1[9], wC[9], wC1[9];

#pragma unroll
    for (int t = 0; t < 9; ++t) {
        float xo = xoff[t], yo = yoff[t];
        float pre   = v0 + xo + yo * (float)WP;
        float ov    = o * xo + yo * (float)WP;       // reference quirk: y applied twice
        float after = pre + ov;
        float avf  = fminf(fmaxf(pre + floorf(ov), 0.0f), (float)(PP - 1));
        float avf1 = fminf(fmaxf(avf + xo,         0.0f), (float)(PP - 1));
        float avc  = fminf(fmaxf(pre + ceilf(ov),  0.0f), (float)(PP - 1));
        float avc1 = fminf(fmaxf(avc + xo,         0.0f), (float)(PP - 1));
        float g1 = fabsf((after - avf)  * (1.0f / (float)WP));
        float g2 = fabsf((avc1 - after) * (1.0f / (float)WP));
        wF[t]  = g1 * fabsf(after - avf);
        wF1[t] = g1 * fabsf(avf1 - after);
        wC1[t] = g2 * fabsf(after - avc1);
        wC[t]  = g2 * fabsf(avc - after);
        iF[t]  = remap_pad(avf,  wF[t]);
        iF1[t] = remap_pad(avf1, wF1[t]);
        iC[t]  = remap_pad(avc,  wC[t]);
        iC1[t] = remap_pad(avc1, wC1[t]);
    }

    for (int ci = 0; ci < 32; ++ci) {
        int c = c0 + ci;
        const float* xc = x + (size_t)c * HW;
        const float* rc = rout + c * 9;              // uniform -> scalar loads
        float acc = 0.0f;
#pragma unroll
        for (int t = 0; t < 9; ++t) {
            float v = wF[t]  * xc[iF[t]]  + wF1[t] * xc[iF1[t]]
                    + wC1[t] * xc[iC1[t]] + wC[t]  * xc[iC[t]];
            acc += v * rc[t];
        }
        y0[(size_t)c * HW + p] = acc;
    }
}

// ---------------------------------------------------------------------------
// Kernel 2: f32 WMMA GEMM  Out[COUT,HW] = Wt[COUT,K] * In[K,HW]
// One wave owns a 64x16 output strip (4 m-tiles) so one B register pair is
// reused by 4 V_WMMA_F32_16X16X4_F32 ops -> 4x fewer activation loads and
// 4 independent accumulators to cover the WMMA->WMMA hazard window.
// ---------------------------------------------------------------------------
template<int K>
__global__ void gemm_wmma_f32(const float* __restrict__ Wt,
                              const float* __restrict__ In,
                              float* __restrict__ Out)
{
    const int MG  = 4;                 // m-tiles per wave
    const int MWG = COUT / (16 * MG);  // 2 wave-groups along M
    int wave = (int)((blockIdx.x * blockDim.x + threadIdx.x) >> 5);
    int lane = threadIdx.x & 31;
    int mg = wave % MWG;
    int nt = wave / MWG;
    int m0 = mg * 16 * MG, n0 = nt * 16;
    int lh = lane >> 4;                // 0/1 (K-pair select)
    int ll = lane & 15;                // M (for A) / N (for B,C)

    // A 16x4 layout: vgpr v, lane l -> M = l&15, K = 2*(l>>4)+v
    // A tiles g=0..3 live at constant offsets g*16*K from one base pointer.
    const float* aptr = Wt + (size_t)(m0 + ll) * K + 2 * lh;
    // B 4x16 layout: vgpr v, lane l -> K = 2*(l>>4)+v, N = l&15
    const float* bptr = In + (size_t)(2 * lh) * HW + n0 + ll;

    v8f acc[MG];
#pragma unroll
    for (int g = 0; g < MG; ++g) acc[g] = (v8f){};

    for (int k = 0; k < K; k += 4) {
        v2f b;
        b.x = bptr[0];
        b.y = bptr[HW];
        bptr += (size_t)4 * HW;
#pragma unroll
        for (int g = 0; g < MG; ++g) {
            v2f a;
            a.x = aptr[g * 16 * K + 0];
            a.y = aptr[g * 16 * K + 1];
            acc[g] = __builtin_amdgcn_wmma_f32_16x16x4_f32(
                         false, a, false, b, (short)0, acc[g], false, false);
        }
        aptr += 4;
    }

    // C/D layout: vgpr r, lane l -> M = r + 8*(l>>4), N = l&15
#pragma unroll
    for (int g = 0; g < MG; ++g) {
        float* op = Out + (size_t)(m0 + g * 16 + 8 * lh) * HW + n0 + ll;
#pragma unroll
        for (int r = 0; r < 8; ++r) op[(size_t)r * HW] = acc[g][r];
    }
}

// ---------------------------------------------------------------------------
// Kernel 3: per-channel BN statistics -> folded scale/shift
// ---------------------------------------------------------------------------
__global__ void bn_stats_kernel(const float* __restrict__ z,
                                const float* __restrict__ g,
                                const float* __restrict__ b,
                                float* __restrict__ stats)   // [0..127]=scale [128..255]=shift
{
    int c = blockIdx.x;
    const float* zc = z + (size_t)c * HW;
    float s = 0.0f, s2 = 0.0f;
    for (int i = threadIdx.x; i < HW; i += 256) {
        float v = zc[i];
        s += v; s2 += v * v;
    }
    __shared__ float sh[256], sh2[256];
    sh[threadIdx.x] = s; sh2[threadIdx.x] = s2;
    __syncthreads();
    for (int off = 128; off > 0; off >>= 1) {
        if ((int)threadIdx.x < off) {
            sh[threadIdx.x]  += sh[threadIdx.x + off];
            sh2[threadIdx.x] += sh2[threadIdx.x + off];
        }
        __syncthreads();
    }
    if (threadIdx.x == 0) {
        float mean = sh[0] * (1.0f / (float)HW);
        float var  = sh2[0] * (1.0f / (float)HW) - mean * mean;
        float rs   = rsqrtf(var + 1e-5f);
        float sc   = g[c] * rs;
        stats[c]        = sc;
        stats[COUT + c] = b[c] - mean * sc;
    }
}

// ---------------------------------------------------------------------------
// Kernel 4: fused BN-apply + LeakyReLU + depthwise 3x3 (zero pad on activations)
// ---------------------------------------------------------------------------
__global__ void dw_bn_act_kernel(const float* __restrict__ z,
                                 const float* __restrict__ stats,
                                 const float* __restrict__ wdw,   // (128,9)
                                 float* __restrict__ out)
{
    int p = blockIdx.x * 256 + threadIdx.x;
    int c = blockIdx.y;
    if (p >= HW) return;
    int h = p / W, w = p - h * W;
    float sc = stats[c], sf = stats[COUT + c];
    const float* zc = z + (size_t)c * HW;
    const float* wc = wdw + c * 9;
    float acc = 0.0f;
#pragma unroll
    for (int dh = -1; dh <= 1; ++dh)
#pragma unroll
        for (int dw = -1; dw <= 1; ++dw) {
            int hh = h + dh, ww = w + dw;
            if (hh >= 0 && hh < H && ww >= 0 && ww < W) {
                float v = zc[hh * W + ww] * sc + sf;
                v = v >= 0.0f ? v : 0.01f * v;
                acc += v * wc[(dh + 1) * 3 + (dw + 1)];
            }
        }
    out[(size_t)c * HW + p] = acc;
}

// ---------------------------------------------------------------------------
// Kernel 5: final BN-apply + LeakyReLU -> d_out
// ---------------------------------------------------------------------------
__global__ void bn_act_out_kernel(const float* __restrict__ z,
                                  const float* __restrict__ stats,
                                  float* __restrict__ out)
{
    int i = blockIdx.x * 256 + threadIdx.x;
    if (i >= COUT * HW) return;
    int c = i / HW;
    float v = z[i] * stats[c] + stats[COUT + c];
    out[i] = v >= 0.0f ? v : 0.01f * v;
}

// ---------------------------------------------------------------------------
extern "C" void kernel_launch(void* const* d_in, const int* in_sizes, int n_in,
                              void* d_out, int out_size, void* d_ws, size_t ws_size,
                              hipStream_t stream) {
    const float* x        = (const float*)d_in[0];   // (256,64,192)
    const float* x_range  = (const float*)d_in[1];   // (1,1,64,192)
    const float* range_o  = (const float*)d_in[2];   // (256,1,3,3)
    const float* w_reduce = (const float*)d_in[3];   // (128,256)
    const float* g_r      = (const float*)d_in[4];
    const float* b_r      = (const float*)d_in[5];
    const float* dw1      = (const float*)d_in[6];   // (128,1,3,3)
    const float* pw1      = (const float*)d_in[7];   // (128,128)
    const float* g1       = (const float*)d_in[8];
    const float* b1       = (const float*)d_in[9];
    const float* dw2      = (const float*)d_in[10];  // (128,1,3,3)
    const float* pw2      = (const float*)d_in[11];  // (128,128)
    const float* g2       = (const float*)d_in[12];
    const float* b2       = (const float*)d_in[13];
    float* out = (float*)d_out;

    float* ws = (float*)d_ws;
    float* y0 = ws;                    // 256*12288 floats
    float* zb = y0 + (size_t)CIN * HW; // 128*12288
    float* db = zb + (size_t)COUT * HW;// 128*12288
    float* st = db + (size_t)COUT * HW;// 256

    // waves = (COUT/64) * (HW/16) = 2 * 768 = 1536 -> 192 blocks x 8 waves
    const int gemm_blocks = (COUT / 64) * (HW / 16) / 8;

    // 1) deformable gather + range_out contraction
    deform_gather_kernel<<<dim3(HW / 256, CIN / 32), 256, 0, stream>>>(x, x_range, range_o, y0);
    // 2) reduce GEMM (K=256) + BN stats
    gemm_wmma_f32<CIN><<<gemm_blocks, 256, 0, stream>>>(w_reduce, y0, zb);
    bn_stats_kernel<<<COUT, 256, 0, stream>>>(zb, g_r, b_r, st);
    // 3) block 1: BN+leaky+dw3x3, pointwise GEMM, stats
    dw_bn_act_kernel<<<dim3(HW / 256, COUT), 256, 0, stream>>>(zb, st, dw1, db);
    gemm_wmma_f32<COUT><<<gemm_blocks, 256, 0, stream>>>(pw1, db, zb);
    bn_stats_kernel<<<COUT, 256, 0, stream>>>(zb, g1, b1, st);
    // 4) block 2
    dw_bn_act_kernel<<<dim3(HW / 256, COUT), 256, 0, stream>>>(zb, st, dw2, db);
    gemm_wmma_f32<COUT><<<gemm_blocks, 256, 0, stream>>>(pw2, db, zb);
    bn_stats_kernel<<<COUT, 256, 0, stream>>>(zb, g2, b2, st);
    // 5) final BN + leaky -> output
    bn_act_out_kernel<<<(COUT * HW) / 256, 256, 0, stream>>>(zb, st, out);
}